// RNNmodel_51642686767750
// MI455X (gfx1250) — compile-verified
//
#include <hip/hip_runtime.h>
#include <hip/hip_bf16.h>

// ---------------------------------------------------------------------------
// EKF forward pass on gfx1250 (MI455X), fp32 WMMA 16x16x4.
// One wave = 16 batch rows; T=128 sequential; per-lane x/P state.
// Weights LDS-resident. 7 fence-stages per timestep: forward process/Q/sensor
// chains run stage-interleaved with the Jacobian software-pipelined one layer
// behind (its masks). A-fragments hoisted => in-place tiles are legal.
// ---------------------------------------------------------------------------

typedef __attribute__((ext_vector_type(2))) float v2f;
typedef __attribute__((ext_vector_type(8))) float v8f;

constexpr int Bn = 4096;
constexpr int Tn = 128;
constexpr int LSTR = 68;            // staging row stride (floats): conflict free
constexpr int BUFF = 16 * LSTR;     // one staging tile
constexpr int NBUF = 11;            // B0,P1..P5,S1,JA,JB,JC,JD

constexpr int padK(int K) { return (K + 3) & ~3; }
constexpr int padN(int N) { return (N + 15) & ~15; }
constexpr int wstr(int N) { return padN(N) == 16 ? 16 : padN(N) + 4; }

// ---- LDS weight arena offsets (floats), all pre-padded/zero-filled --------
constexpr int OFF_PW1  = 0;
constexpr int OFF_PB1  = OFF_PW1  + padK(4)  * wstr(32);
constexpr int OFF_PW1A = OFF_PB1  + padN(32);
constexpr int OFF_PB1A = OFF_PW1A + padK(32) * wstr(32);
constexpr int OFF_PW2  = OFF_PB1A + padN(32);
constexpr int OFF_PB2  = OFF_PW2  + padK(32) * wstr(64);
constexpr int OFF_PW2A = OFF_PB2  + padN(64);
constexpr int OFF_PB2A = OFF_PW2A + padK(64) * wstr(64);
constexpr int OFF_PW3  = OFF_PB2A + padN(64);
constexpr int OFF_PB3  = OFF_PW3  + padK(64) * wstr(4);
constexpr int OFF_QW1  = OFF_PB3  + padN(4);
constexpr int OFF_QB1  = OFF_QW1  + padK(4)  * wstr(32);
constexpr int OFF_QW1A = OFF_QB1  + padN(32);
constexpr int OFF_QB1A = OFF_QW1A + padK(32) * wstr(64);
constexpr int OFF_QW2  = OFF_QB1A + padN(64);
constexpr int OFF_QB2  = OFF_QW2  + padK(64) * wstr(64);
constexpr int OFF_QW3  = OFF_QB2  + padN(64);
constexpr int OFF_QB3  = OFF_QW3  + padK(64) * wstr(4);
constexpr int OFF_SW1  = OFF_QB3  + padN(4);
constexpr int OFF_SB1  = OFF_SW1  + padK(2)  * wstr(32);
constexpr int OFF_SW1A = OFF_SB1  + padN(32);
constexpr int OFF_SB1A = OFF_SW1A + padK(32) * wstr(64);
constexpr int OFF_SW2  = OFF_SB1A + padN(64);
constexpr int OFF_SB2  = OFF_SW2  + padK(64) * wstr(64);
constexpr int OFF_HW   = OFF_SB2  + padN(64);           // combined [sw3|rw1]
constexpr int OFF_HB   = OFF_HW   + padK(64) * wstr(4); // combined [sb3|rb1]
constexpr int WTOT     = OFF_HB   + padN(4);            // ~24.8K floats

constexpr int SMEM_FLOATS = WTOT + NBUF * BUFF;         // ~147 KB total

struct EkfParams {
  const float* raw_obs;            // [B,T,2]
  const float* init_state;         // [B,4]
  const float *pw1, *pb1, *pw1a, *pb1a, *pw2, *pb2, *pw2a, *pb2a, *pw3, *pb3;
  const float *qw1, *qb1, *qw1a, *qb1a, *qw2, *qb2, *qw3, *qb3, *q_learned, *q_fixed;
  const float *sw1, *sb1, *sw1a, *sb1a, *sw2, *sb2, *sw3, *sb3;
  const float *rw1, *rb1, *r_learned, *r_fixed;
};

// Wave-private LDS fence: all our DS stores complete before subsequent reads.
__device__ __forceinline__ void wave_lds_fence() {
  __builtin_amdgcn_wave_barrier();
  asm volatile("s_wait_dscnt 0" ::: "memory");
  __builtin_amdgcn_wave_barrier();
}

// Cooperative, zero-padded copies global -> LDS weight arena
template <int K, int N>
__device__ __forceinline__ void load_mat(float* dst, const float* src, int tid) {
  constexpr int KP = padK(K), STR = wstr(N);
  for (int idx = tid; idx < KP * STR; idx += 32) {
    const int k = idx / STR, n = idx % STR;
    dst[idx] = (k < K && n < N) ? src[k * N + n] : 0.0f;
  }
}
template <int N>
__device__ __forceinline__ void load_vec(float* dst, const float* src, int tid) {
  for (int idx = tid; idx < padN(N); idx += 32)
    dst[idx] = (idx < N) ? src[idx] : 0.0f;
}
// combined head: cols 0-1 = sw3 (z), cols 2-3 = rw1 (R), rest 0
__device__ __forceinline__ void load_head(float* dstw, float* dstb,
                                          const float* sw3, const float* rw1,
                                          const float* sb3, const float* rb1,
                                          int tid) {
  for (int idx = tid; idx < 64 * 16; idx += 32) {
    const int k = idx / 16, n = idx % 16;
    float v = 0.0f;
    if (n < 2) v = sw3[k * 2 + n];
    else if (n < 4) v = rw1[k * 2 + (n - 2)];
    dstw[idx] = v;
  }
  if (tid < 16) {
    float v = 0.0f;
    if (tid < 2) v = sb3[tid];
    else if (tid < 4) v = rb1[tid - 2];
    dstb[tid] = v;
  }
}

// One dense layer on a 16-row batch tile via V_WMMA_F32_16X16X4_F32.
// A-fragments hoisted before any store => in-place (out aliasing in) is safe.
template <int K, int N, bool RELU, bool MASKED, bool HASBIAS>
__device__ __forceinline__ void layer16(const float* in, const float* w,
                                        const float* bias, const float* mact,
                                        float* out, int lane) {
  constexpr int KP = padK(K);
  constexpr int NT = padN(N) / 16;
  constexpr int WS = wstr(N);
  const int col   = lane & 15;            // D/B column owned by this lane
  const int hi    = lane >> 4;            // half-wave id
  const int koff  = hi * 2;               // A/B: VGPR0 holds K=0 (lo) / K=2 (hi)
  const int rbase = hi * 8;               // D rows: M = v + 8*hi

  v2f afrag[KP / 4];
#pragma unroll
  for (int kt = 0; kt < KP / 4; ++kt)
    afrag[kt] = *reinterpret_cast<const v2f*>(&in[col * LSTR + kt * 4 + koff]);

#pragma unroll
  for (int nt = 0; nt < NT; ++nt) {
    const int n = nt * 16 + col;
    const float bv = HASBIAS ? bias[n] : 0.0f;
    v8f c;
#pragma unroll
    for (int v = 0; v < 8; ++v) c[v] = bv;   // bias depends on column only

#pragma unroll
    for (int kt = 0; kt < KP / 4; ++kt) {
      const int k0 = kt * 4 + koff;
      v2f b;
      b.x = w[(k0)     * WS + n];
      b.y = w[(k0 + 1) * WS + n];
      c = __builtin_amdgcn_wmma_f32_16x16x4_f32(false, afrag[kt], false, b,
                                                (short)0, c, false, false);
    }
#pragma unroll
    for (int v = 0; v < 8; ++v) {
      float d = c[v];
      if (RELU) d = fmaxf(d, 0.0f);
      const int r = rbase + v;
      if (MASKED) d = (mact[r * LSTR + nt * 16 + col] > 0.0f) ? d : 0.0f;
      out[r * LSTR + nt * 16 + col] = d;
    }
  }
}

__launch_bounds__(32)
__global__ void ekf_wmma_kernel(EkfParams prm, float* __restrict__ out) {
  __shared__ float smem[SMEM_FLOATS];

  const int tid  = threadIdx.x;
  const int lane = tid & 31;
  const int row  = lane & 15;
  const int col  = lane & 15;
  const int hi   = lane >> 4;
  const int rbase = hi * 8;

  float* wl = smem;                       // weight arena
  load_mat<4, 32>(wl + OFF_PW1,  prm.pw1,  tid);
  load_vec<32>   (wl + OFF_PB1,  prm.pb1,  tid);
  load_mat<32,32>(wl + OFF_PW1A, prm.pw1a, tid);
  load_vec<32>   (wl + OFF_PB1A, prm.pb1a, tid);
  load_mat<32,64>(wl + OFF_PW2,  prm.pw2,  tid);
  load_vec<64>   (wl + OFF_PB2,  prm.pb2,  tid);
  load_mat<64,64>(wl + OFF_PW2A, prm.pw2a, tid);
  load_vec<64>   (wl + OFF_PB2A, prm.pb2a, tid);
  load_mat<64, 4>(wl + OFF_PW3,  prm.pw3,  tid);
  load_vec<4>    (wl + OFF_PB3,  prm.pb3,  tid);
  load_mat<4, 32>(wl + OFF_QW1,  prm.qw1,  tid);
  load_vec<32>   (wl + OFF_QB1,  prm.qb1,  tid);
  load_mat<32,64>(wl + OFF_QW1A, prm.qw1a, tid);
  load_vec<64>   (wl + OFF_QB1A, prm.qb1a, tid);
  load_mat<64,64>(wl + OFF_QW2,  prm.qw2,  tid);
  load_vec<64>   (wl + OFF_QB2,  prm.qb2,  tid);
  load_mat<64, 4>(wl + OFF_QW3,  prm.qw3,  tid);
  load_vec<4>    (wl + OFF_QB3,  prm.qb3,  tid);
  load_mat<2, 32>(wl + OFF_SW1,  prm.sw1,  tid);
  load_vec<32>   (wl + OFF_SB1,  prm.sb1,  tid);
  load_mat<32,64>(wl + OFF_SW1A, prm.sw1a, tid);
  load_vec<64>   (wl + OFF_SB1A, prm.sb1a, tid);
  load_mat<64,64>(wl + OFF_SW2,  prm.sw2,  tid);
  load_vec<64>   (wl + OFF_SB2,  prm.sb2,  tid);
  load_head(wl + OFF_HW, wl + OFF_HB, prm.sw3, prm.rw1, prm.sb3, prm.rb1, tid);
  __syncthreads();

  float* arena = smem + WTOT;
  float* B0 = arena + 0  * BUFF;   // x staging + whole Q chain (in-place)
  float* P1 = arena + 1  * BUFF;   // h1 mask; later t2[e0], jd packed
  float* P2 = arena + 2  * BUFF;   // h1a mask; later t2[e1]
  float* P3 = arena + 3  * BUFF;   // h2 mask
  float* P4 = arena + 4  * BUFF;   // h2a mask
  float* P5 = arena + 5  * BUFF;   // delta
  float* S1 = arena + 6  * BUFF;   // obs + whole sensor chain (in-place)
  float* JA = arena + 7  * BUFF;   // t1/t1a e0,e1 (packed halves)
  float* JB = arena + 8  * BUFF;   // t1/t1a e2,e3
  float* JC = arena + 9  * BUFF;   // t2[e2]/t2a[e2]
  float* JD = arena + 10 * BUFF;   // t2[e3]/t2a[e3]

  const int b = blockIdx.x * 16 + row;
  const bool writer = (lane < 16);   // lanes 16..31 mirror state

  float x[4];
#pragma unroll
  for (int i = 0; i < 4; ++i) x[i] = prm.init_state[b * 4 + i];
  float P[4][4];
#pragma unroll
  for (int i = 0; i < 4; ++i)
#pragma unroll
    for (int j = 0; j < 4; ++j) P[i][j] = (i == j) ? 1.0f : 0.0f;

  float ql[4], qf[4];
#pragma unroll
  for (int i = 0; i < 4; ++i) { ql[i] = prm.q_learned[i]; qf[i] = prm.q_fixed[i]; }
  float rl[2], rf[2];
#pragma unroll
  for (int i = 0; i < 2; ++i) { rl[i] = prm.r_learned[i]; rf[i] = prm.r_fixed[i]; }

#pragma unroll 1
  for (int t = 0; t < Tn; ++t) {
    const v2f obs = *reinterpret_cast<const v2f*>(&prm.raw_obs[(b * Tn + t) * 2]);

    // ---- stage 0: stage x (process+Q) and obs (sensor) ---------------------
    if (writer) {
#pragma unroll
      for (int k = 0; k < 4; ++k) B0[row * LSTR + k] = x[k];
      S1[row * LSTR + 0] = obs.x;
      S1[row * LSTR + 1] = obs.y;
      S1[row * LSTR + 2] = 0.0f;
      S1[row * LSTR + 3] = 0.0f;
    }
    wave_lds_fence();

    // ---- stage 1 -----------------------------------------------------------
    layer16<4, 32, true, false, true>(B0, wl + OFF_PW1, wl + OFF_PB1, nullptr, P1, lane);
    layer16<4, 32, true, false, true>(B0, wl + OFF_QW1, wl + OFF_QB1, nullptr, B0, lane);
    layer16<2, 32, true, false, true>(S1, wl + OFF_SW1, wl + OFF_SB1, nullptr, S1, lane);
    wave_lds_fence();

    // ---- stage 2 (+ J1 elementwise: t1_e = pw1[e,:] * relu'(h1)) -----------
    layer16<32,32, true, false, true>(P1, wl + OFF_PW1A, wl + OFF_PB1A, nullptr, P2, lane);
    layer16<32,64, true, false, true>(B0, wl + OFF_QW1A, wl + OFF_QB1A, nullptr, B0, lane);
    layer16<32,64, true, false, true>(S1, wl + OFF_SW1A, wl + OFF_SB1A, nullptr, S1, lane);
    {
      constexpr int WS = wstr(32);
#pragma unroll
      for (int nt = 0; nt < 2; ++nt) {
        const int n = nt * 16 + col;
        const float w0 = wl[OFF_PW1 + 0 * WS + n];
        const float w1 = wl[OFF_PW1 + 1 * WS + n];
        const float w2 = wl[OFF_PW1 + 2 * WS + n];
        const float w3 = wl[OFF_PW1 + 3 * WS + n];
#pragma unroll
        for (int v = 0; v < 8; ++v) {
          const int r = rbase + v;
          const bool m = P1[r * LSTR + n] > 0.0f;
          JA[r * LSTR + 0  + n] = m ? w0 : 0.0f;
          JA[r * LSTR + 32 + n] = m ? w1 : 0.0f;
          JB[r * LSTR + 0  + n] = m ? w2 : 0.0f;
          JB[r * LSTR + 32 + n] = m ? w3 : 0.0f;
        }
      }
    }
    wave_lds_fence();

    // ---- stage 3 (+ J2 in-place on packed halves) --------------------------
    layer16<32,64, true, false, true>(P2, wl + OFF_PW2, wl + OFF_PB2, nullptr, P3, lane);
    layer16<64,64, true, false, true>(B0, wl + OFF_QW2, wl + OFF_QB2, nullptr, B0, lane);
    layer16<64,64, true, false, true>(S1, wl + OFF_SW2, wl + OFF_SB2, nullptr, S1, lane); // enc
    layer16<32,32, false, true, false>(JA,      wl + OFF_PW1A, nullptr, P2, JA,      lane);
    layer16<32,32, false, true, false>(JA + 32, wl + OFF_PW1A, nullptr, P2, JA + 32, lane);
    layer16<32,32, false, true, false>(JB,      wl + OFF_PW1A, nullptr, P2, JB,      lane);
    layer16<32,32, false, true, false>(JB + 32, wl + OFF_PW1A, nullptr, P2, JB + 32, lane);
    wave_lds_fence();

    // ---- stage 4 (+ J3; P1,P2 masks dead -> become t2 tiles) ---------------
    layer16<64,64, true, false, true>(P3, wl + OFF_PW2A, wl + OFF_PB2A, nullptr, P4, lane);
    layer16<64, 4, false,false, true>(B0, wl + OFF_QW3,  wl + OFF_QB3,  nullptr, B0, lane); // q pre
    layer16<64, 4, false,false, true>(S1, wl + OFF_HW,   wl + OFF_HB,   nullptr, S1, lane); // z|R
    layer16<32,64, false, true, false>(JA,      wl + OFF_PW2, nullptr, P3, P1, lane);
    layer16<32,64, false, true, false>(JA + 32, wl + OFF_PW2, nullptr, P3, P2, lane);
    layer16<32,64, false, true, false>(JB,      wl + OFF_PW2, nullptr, P3, JC, lane);
    layer16<32,64, false, true, false>(JB + 32, wl + OFF_PW2, nullptr, P3, JD, lane);
    wave_lds_fence();

    // ---- stage 5 (+ J4 in-place) -------------------------------------------
    layer16<64, 4, false,false, true>(P4, wl + OFF_PW3, wl + OFF_PB3, nullptr, P5, lane); // delta
    layer16<64,64, false, true, false>(P1, wl + OFF_PW2A, nullptr, P4, P1, lane);
    layer16<64,64, false, true, false>(P2, wl + OFF_PW2A, nullptr, P4, P2, lane);
    layer16<64,64, false, true, false>(JC, wl + OFF_PW2A, nullptr, P4, JC, lane);
    layer16<64,64, false, true, false>(JD, wl + OFF_PW2A, nullptr, P4, JD, lane);
    wave_lds_fence();

    // ---- stage 6: J5, packed into P1 at col offsets e*16 -------------------
    layer16<64, 4, false, false, false>(P1, wl + OFF_PW3, nullptr, nullptr, P1,      lane);
    layer16<64, 4, false, false, false>(P2, wl + OFF_PW3, nullptr, nullptr, P1 + 16, lane);
    layer16<64, 4, false, false, false>(JC, wl + OFF_PW3, nullptr, nullptr, P1 + 32, lane);
    layer16<64, 4, false, false, false>(JD, wl + OFF_PW3, nullptr, nullptr, P1 + 48, lane);
    wave_lds_fence();

    // ---- extraction --------------------------------------------------------
    float xp[4], qd[4], rd[2];
#pragma unroll
    for (int i = 0; i < 4; ++i) xp[i] = x[i] + P5[row * LSTR + i];
#pragma unroll
    for (int i = 0; i < 4; ++i) {
      float d = B0[row * LSTR + i] + ql[i];
      qd[i] = d * d + qf[i];
    }
    const float z0 = S1[row * LSTR + 0];
    const float z1 = S1[row * LSTR + 1];
#pragma unroll
    for (int i = 0; i < 2; ++i) {
      float d = fmaxf(S1[row * LSTR + 2 + i], 0.0f) + rl[i];  // R head ReLU here
      rd[i] = d * d + rf[i];
    }
    float F[4][4];
#pragma unroll
    for (int e = 0; e < 4; ++e)
#pragma unroll
      for (int i = 0; i < 4; ++i)
        F[i][e] = P1[row * LSTR + e * 16 + i] + ((i == e) ? 1.0f : 0.0f);

    // ---- per-lane Kalman update (fp32 scalar) ------------------------------
    float FP[4][4];
#pragma unroll
    for (int i = 0; i < 4; ++i)
#pragma unroll
      for (int j = 0; j < 4; ++j) {
        float s = 0.0f;
#pragma unroll
        for (int k = 0; k < 4; ++k) s += F[i][k] * P[k][j];
        FP[i][j] = s;
      }
    float Pp[4][4];
#pragma unroll
    for (int i = 0; i < 4; ++i)
#pragma unroll
      for (int j = 0; j < 4; ++j) {
        float s = 0.0f;
#pragma unroll
        for (int k = 0; k < 4; ++k) s += FP[i][k] * F[j][k];
        Pp[i][j] = s;
      }
#pragma unroll
    for (int i = 0; i < 4; ++i) Pp[i][i] += qd[i];

    const float y0 = z0 - xp[0];
    const float y1 = z1 - xp[1];
    const float s00 = Pp[0][0] + rd[0];
    const float s01 = Pp[0][1];
    const float s10 = Pp[1][0];
    const float s11 = Pp[1][1] + rd[1];
    const float inv = 1.0f / (s00 * s11 - s01 * s10);

    float Kg[4][2];
#pragma unroll
    for (int i = 0; i < 4; ++i) {
      Kg[i][0] = (Pp[i][0] * s11 - Pp[i][1] * s01) * inv;
      Kg[i][1] = (Pp[i][1] * s00 - Pp[i][0] * s10) * inv;
    }
#pragma unroll
    for (int i = 0; i < 4; ++i) x[i] = xp[i] + Kg[i][0] * y0 + Kg[i][1] * y1;
#pragma unroll
    for (int i = 0; i < 4; ++i)
#pragma unroll
      for (int j = 0; j < 4; ++j)
        P[i][j] = Pp[i][j] - Kg[i][0] * Pp[0][j] - Kg[i][1] * Pp[1][j];

    if (writer) {
      float4 o = make_float4(x[0], x[1], x[2], x[3]);
      *reinterpret_cast<float4*>(&out[(b * Tn + t) * 4]) = o;
    }
  }
}

extern "C" void kernel_launch(void* const* d_in, const int* in_sizes, int n_in,
                              void* d_out, int out_size, void* d_ws, size_t ws_size,
                              hipStream_t stream) {
  (void)in_sizes; (void)out_size; (void)d_ws; (void)ws_size;
  if (n_in < 34) return;

  EkfParams p;
  p.raw_obs    = (const float*)d_in[0];
  p.init_state = (const float*)d_in[1];
  p.pw1  = (const float*)d_in[2];  p.pb1  = (const float*)d_in[3];
  p.pw1a = (const float*)d_in[4];  p.pb1a = (const float*)d_in[5];
  p.pw2  = (const float*)d_in[6];  p.pb2  = (const float*)d_in[7];
  p.pw2a = (const float*)d_in[8];  p.pb2a = (const float*)d_in[9];
  p.pw3  = (const float*)d_in[10]; p.pb3  = (const float*)d_in[11];
  p.qw1  = (const float*)d_in[12]; p.qb1  = (const float*)d_in[13];
  p.qw1a = (const float*)d_in[14]; p.qb1a = (const float*)d_in[15];
  p.qw2  = (const float*)d_in[16]; p.qb2  = (const float*)d_in[17];
  p.qw3  = (const float*)d_in[18]; p.qb3  = (const float*)d_in[19];
  p.q_learned = (const float*)d_in[20]; p.q_fixed = (const float*)d_in[21];
  p.sw1  = (const float*)d_in[22]; p.sb1  = (const float*)d_in[23];
  p.sw1a = (const float*)d_in[24]; p.sb1a = (const float*)d_in[25];
  p.sw2  = (const float*)d_in[26]; p.sb2  = (const float*)d_in[27];
  p.sw3  = (const float*)d_in[28]; p.sb3  = (const float*)d_in[29];
  p.rw1  = (const float*)d_in[30]; p.rb1  = (const float*)d_in[31];
  p.r_learned = (const float*)d_in[32]; p.r_fixed = (const float*)d_in[33];

  dim3 grid(Bn / 16);     // 256 blocks
  dim3 block(32);         // 1 wave
  ekf_wmma_kernel<<<grid, block, 0, stream>>>(p, (float*)d_out);
}